// IouLoss_26766236189166
// MI455X (gfx1250) — compile-verified
//
#include <hip/hip_runtime.h>
#include <math.h>

#define Cdim 10
#define Hdim 256
#define Wdim 256
#define Kdim 500
#define NBOX (32 * Kdim)
#define EPSF 1e-7f

struct V2f { float x, y; };

__device__ inline float shflf(float v, int src) { return __shfl(v, src, 32); }

// ---- _cal_box: tt=p[0:2], rr=p[2:4], bb=p[4:6], ll=p[6:8]; cen=(tt+bb)/2 ----
__device__ inline void cal_box(const float p[8], V2f out[4]) {
    float cx = (p[0] + p[4]) * 0.5f;
    float cy = (p[1] + p[5]) * 0.5f;
    out[0].x = (p[0] + p[2]) - cx; out[0].y = (p[1] + p[3]) - cy; // tt+rr
    out[1].x = (p[4] + p[2]) - cx; out[1].y = (p[5] + p[3]) - cy; // bb+rr
    out[2].x = (p[4] + p[6]) - cx; out[2].y = (p[5] + p[7]) - cy; // bb+ll
    out[3].x = (p[0] + p[6]) - cx; out[3].y = (p[1] + p[7]) - cy; // tt+ll
}

// stable ascending rank sort of 4 points by precomputed angles; register-only
__device__ inline void sort4_by_ang(V2f pts[4], const float ang[4]) {
    V2f out[4];
    #pragma unroll
    for (int i = 0; i < 4; ++i) {
        int r = 0;
        #pragma unroll
        for (int j = 0; j < 4; ++j)
            r += (ang[j] < ang[i]) || ((ang[j] == ang[i]) && (j < i));
        #pragma unroll
        for (int k = 0; k < 4; ++k)
            if (r == k) out[k] = pts[i];          // static index -> v_cndmask, no scratch
    }
    #pragma unroll
    for (int i = 0; i < 4; ++i) pts[i] = out[i];
}

__device__ inline float shoelace4(const V2f p[4]) {
    float s = 0.0f;
    #pragma unroll
    for (int i = 0; i < 4; ++i) {
        int j = (i + 1) & 3;
        s += p[i].x * p[j].y - p[j].x * p[i].y;
    }
    return 0.5f * fabsf(s);
}

// Single workgroup. Phase 1: 32 waves scan reg_mask (float4 / b128 loads) for the
// last active slot. Phase 2: wave 0 runs the whole polygon-clip epilogue with one
// candidate point per lane (ballot_w32 / shfl butterflies / ds_permute_b32).
__global__ void __launch_bounds__(1024)
iou_last_loss_kernel(const float* __restrict__ pred_wh,
                     const float* __restrict__ wh_target,
                     const float* __restrict__ reg_mask,
                     const int*   __restrict__ ind,
                     float* __restrict__ out) {
    __shared__ int smax[32];
    const int tid = threadIdx.x;

    // gfx1250: global_prefetch_b8 — start pulling the mask range
    __builtin_prefetch(&reg_mask[tid * 4], 0, 0);
    __builtin_prefetch(&reg_mask[tid * 4 + 8192], 0, 0);

    // ---- Phase 1: find last index with mask > 0 (vectorized b128 scan) ----
    int best = -1;
    const float4* m4 = reinterpret_cast<const float4*>(reg_mask);
    for (int vi = tid; vi < NBOX / 4; vi += 1024) {
        float4 m = m4[vi];
        int base = vi << 2;
        if (m.x > 0.0f) best = base;        // vi increases -> last write wins
        if (m.y > 0.0f) best = base + 1;
        if (m.z > 0.0f) best = base + 2;
        if (m.w > 0.0f) best = base + 3;
    }
    #pragma unroll
    for (int off = 16; off > 0; off >>= 1) {
        int o = __shfl_xor(best, off, 32);
        best = (o > best) ? o : best;
    }
    if ((tid & 31) == 0) smax[tid >> 5] = best;
    __syncthreads();
    if (tid >= 32) return;

    // ---- Phase 2: wave 0 only ----
    const int lane = tid;
    int lv = smax[lane];
    #pragma unroll
    for (int off = 16; off > 0; off >>= 1) {
        int o = __shfl_xor(lv, off, 32);
        lv = (o > lv) ? o : lv;
    }
    const int last = lv;                    // uniform across the wave

    float loss = 0.0f;
    if (last >= 0) {
        const int bi  = last / Kdim;
        const int pos = ind[last];

        // lane-parallel gather of the 8+8 coordinates, broadcast by shuffle
        float av = 0.0f, gv = 0.0f;
        if (lane < 8) {
            av = pred_wh[(size_t)bi * (size_t)(Cdim * Hdim * Wdim)
                         + (size_t)lane * (size_t)(Hdim * Wdim) + (size_t)pos];
            gv = wh_target[(size_t)last * Cdim + lane];
        }
        float a[8], g[8];
        #pragma unroll
        for (int c = 0; c < 8; ++c) { a[c] = shflf(av, c); g[c] = shflf(gv, c); }

        // boxes (replicated on all lanes; cheap)
        V2f A[4], Bq[4];
        cal_box(a, A);
        cal_box(g, Bq);

        // CCW sort: 8 atan2f computed one-per-lane, gathered by shuffle
        float cxA = (A[0].x + A[1].x + A[2].x + A[3].x) * 0.25f;
        float cyA = (A[0].y + A[1].y + A[2].y + A[3].y) * 0.25f;
        float cxB = (Bq[0].x + Bq[1].x + Bq[2].x + Bq[3].x) * 0.25f;
        float cyB = (Bq[0].y + Bq[1].y + Bq[2].y + Bq[3].y) * 0.25f;
        V2f  ps  = (lane < 4) ? A[lane & 3] : Bq[lane & 3];
        float pcx = (lane < 4) ? cxA : cxB;
        float pcy = (lane < 4) ? cyA : cyB;
        float sang = atan2f(ps.y - pcy, ps.x - pcx);
        float angA[4], angB[4];
        #pragma unroll
        for (int i = 0; i < 4; ++i) { angA[i] = shflf(sang, i); angB[i] = shflf(sang, 4 + i); }
        sort4_by_ang(A, angA);
        sort4_by_ang(Bq, angB);

        V2f An[4], Bn[4];
        #pragma unroll
        for (int i = 0; i < 4; ++i) { An[i] = A[(i + 1) & 3]; Bn[i] = Bq[(i + 1) & 3]; }

        // ---- one candidate point per lane (0..23), order matches reference ----
        V2f  P; P.x = 0.0f; P.y = 0.0f;
        bool ok = false;
        if (lane < 4) {                       // A vertices inside B
            P = A[lane];
            ok = true;
            #pragma unroll
            for (int j = 0; j < 4; ++j) {
                float ex = Bn[j].x - Bq[j].x, ey = Bn[j].y - Bq[j].y;
                float px = P.x - Bq[j].x,     py = P.y - Bq[j].y;
                ok = ok && (ex * py - ey * px >= -EPSF);
            }
        } else if (lane < 8) {                // B vertices inside A
            P = Bq[lane - 4];
            ok = true;
            #pragma unroll
            for (int j = 0; j < 4; ++j) {
                float ex = An[j].x - A[j].x, ey = An[j].y - A[j].y;
                float px = P.x - A[j].x,     py = P.y - A[j].y;
                ok = ok && (ex * py - ey * px >= -EPSF);
            }
        } else if (lane < 24) {               // edge i of A vs edge j of B
            int li = (lane - 8) >> 2, lj = (lane - 8) & 3;
            float d1x = An[li].x - A[li].x,  d1y = An[li].y - A[li].y;
            float d2x = Bn[lj].x - Bq[lj].x, d2y = Bn[lj].y - Bq[lj].y;
            float den = d1x * d2y - d1y * d2x;
            float dx  = Bq[lj].x - A[li].x,  dy = Bq[lj].y - A[li].y;
            float safe = (fabsf(den) > EPSF) ? den : 1.0f;
            float t = (dx * d2y - dy * d2x) / safe;
            float u = (dx * d1y - dy * d1x) / safe;
            ok = (fabsf(den) > EPSF) &&
                 (t >= -EPSF) && (t <= 1.0f + EPSF) &&
                 (u >= -EPSF) && (u <= 1.0f + EPSF);
            P.x = A[li].x + t * d1x;
            P.y = A[li].y + t * d1y;
        }

        float inter = 0.0f;
        unsigned int bal = __builtin_amdgcn_ballot_w32(ok);   // uniform value
        if (bal != 0u) {
            int fv = __builtin_ctz(bal);
            int nv = __builtin_popcount(bal);

            // centroid over valid candidates (butterfly sum)
            float sxc = ok ? P.x : 0.0f;
            float syc = ok ? P.y : 0.0f;
            #pragma unroll
            for (int off = 16; off > 0; off >>= 1) {
                sxc += __shfl_xor(sxc, off, 32);
                syc += __shfl_xor(syc, off, 32);
            }
            float invn = 1.0f / (float)nv;
            float ccx = sxc * invn, ccy = syc * invn;

            // replace invalid candidates by the first valid one (reference semantics)
            float fx = shflf(P.x, fv), fy = shflf(P.y, fv);
            if (!ok) { P.x = fx; P.y = fy; }

            // angle about centroid, parallel across lanes
            float ang = atan2f(P.y - ccy, P.x - ccx);

            // stable ascending rank among the 24 candidates
            int r = 0;
            #pragma unroll
            for (int j = 0; j < 24; ++j) {
                float aj = shflf(ang, j);
                r += (aj < ang) || ((aj == ang) && (j < lane));
            }
            if (lane >= 24) r = lane;         // park unused lanes on themselves

            // inverse permutation in one instruction: ds_permute_b32 scatter
            float Sx = __int_as_float(__builtin_amdgcn_ds_permute(r << 2, __float_as_int(P.x)));
            float Sy = __int_as_float(__builtin_amdgcn_ds_permute(r << 2, __float_as_int(P.y)));

            // shoelace over the sorted 24-gon
            int   nk = (lane < 24) ? ((lane + 1 == 24) ? 0 : lane + 1) : lane;
            float nx = shflf(Sx, nk), ny = shflf(Sy, nk);
            float cr = (lane < 24) ? (Sx * ny - nx * Sy) : 0.0f;
            #pragma unroll
            for (int off = 16; off > 0; off >>= 1) cr += __shfl_xor(cr, off, 32);
            inter = 0.5f * fabsf(cr);
        }

        // ---- tail (replicated; 6 atanf done one-per-lane) ----
        float uni = shoelace4(A) + shoelace4(Bq) - inter;
        float iou = (uni > 0.0f) ? (inter / uni) : 0.0f;

        float w  = sqrtf((a[2]-a[6])*(a[2]-a[6]) + (a[3]-a[7])*(a[3]-a[7]));
        float h  = sqrtf((a[0]-a[4])*(a[0]-a[4]) + (a[1]-a[5])*(a[1]-a[5]));
        // NOTE: reference uses a[7] here (not b[7]) — replicated intentionally.
        float wt = sqrtf((g[2]-g[6])*(g[2]-g[6]) + (g[3]-a[7])*(g[3]-a[7]));
        float ht = sqrtf((g[0]-g[4])*(g[0]-g[4]) + (g[1]-g[5])*(g[1]-g[5]));

        float th   = (a[1]-a[5]) / (a[0]-a[4]);
        float tth  = (g[1]-g[5]) / (g[0]-g[4]);
        float th1  = (a[3]-a[7]) / (a[2]-a[6]);
        float tth1 = (g[3]-g[7]) / (g[2]-g[6]);

        float xin = 0.0f;
        if      (lane == 0) xin = th;
        else if (lane == 1) xin = tth;
        else if (lane == 2) xin = th1;
        else if (lane == 3) xin = tth1;
        else if (lane == 4) xin = wt / ht;
        else if (lane == 5) xin = w / h;
        float at = atanf(xin);
        float at0 = shflf(at, 0), at1 = shflf(at, 1), at2 = shflf(at, 2);
        float at3 = shflf(at, 3), at4 = shflf(at, 4), at5 = shflf(at, 5);

        float d0 = at0 - at1;
        float d1 = at2 - at3;
        float n  = fminf(d0 * d0, d1 * d1);
        const float c4 = (float)(4.0 / (3.14159265358979323846 * 3.14159265358979323846));
        float dv = at4 - at5;
        float v  = c4 * dv * dv;
        float s  = c4 * n;
        float alpha = (v + s) / (1.0f - iou + v + s);
        loss = alpha * (v + 0.7f * s);
    }

    if (lane == 0) out[0] = loss;
}

extern "C" void kernel_launch(void* const* d_in, const int* in_sizes, int n_in,
                              void* d_out, int out_size, void* d_ws, size_t ws_size,
                              hipStream_t stream) {
    const float* pred_wh   = (const float*)d_in[0];
    const float* wh_target = (const float*)d_in[1];
    const float* reg_mask  = (const float*)d_in[2];
    const int*   ind       = (const int*)d_in[3];
    float* out = (float*)d_out;
    (void)in_sizes; (void)n_in; (void)out_size; (void)d_ws; (void)ws_size;

    iou_last_loss_kernel<<<1, 1024, 0, stream>>>(pred_wh, wh_target, reg_mask, ind, out);
}